// GroupedGraphAttentionNet_86964497809447
// MI455X (gfx1250) — compile-verified
//
#include <hip/hip_runtime.h>
#include <hip/hip_bf16.h>

// CDNA5 / gfx1250 fused GroupedGraphAttentionNet forward.
// All GEMM-shaped stages run on V_WMMA_F32_16X16X4_F32 (exact fp32 matrix op).
typedef float v2f __attribute__((ext_vector_type(2)));
typedef float v8f __attribute__((ext_vector_type(8)));

#define WMMA4(A, Bf, C) \
  __builtin_amdgcn_wmma_f32_16x16x4_f32(false, (A), false, (Bf), (short)0, (C), false, false)

// ---- problem constants (from reference) ----
#define BATCH      524288
#define DIN        64
#define H1D        128
#define H2D        32
#define NG         8
#define NE         4
#define NA         8
#define NH         2

#define SPB        64          // samples per block (4 waves x 16-row M-tiles)
#define NBLK       (BATCH / SPB)
#define NTHR       128

// ---- dynamic-LDS layout (float offsets) ----
#define OFF_W1    0                      // [16k][8n][32 lane][2] frag-major  (8192)
#define OFF_W2    (OFF_W1 + 8192)        // [32k][2n][32][2]                  (4096)
#define OFF_ENCW  (OFF_W2 + 4096)        // folded block-diag, frag-major     (2048)
#define OFF_B1    (OFF_ENCW + 2048)      // (128)
#define OFF_B2    (OFF_B1 + 128)         // (32)
#define OFF_ENCB  (OFF_B2 + 32)          // folded bias (32)
#define OFF_SRCW  (OFF_ENCB + 32)        // frag, pre-scaled by 1/sqrt(A) (64)
#define OFF_DSTW  (OFF_SRCW + 64)        // frag (64)
#define OFF_SRCB  (OFF_DSTW + 64)        // pre-scaled (16)
#define OFF_DSTB  (OFF_SRCB + 16)        // (16)
#define OFF_DECW  (OFF_DSTB + 16)        // (192)
#define OFF_DECB  (OFF_DECW + 192)       // (2) -> pad to even
#define OFF_SLAB  (OFF_DECB + 4)         // per-wave union: hid1[16*132] OR src[128*17]+dst[128*17]
#define SLAB_W    4352                   // per wave
#define OFF_XHAT  (OFF_SLAB + 4 * SLAB_W) // per wave [16][34]
#define OFF_H     (OFF_XHAT + 4 * 544)    // per wave [16][34]
#define OFF_ATT1  (OFF_H + 4 * 544)       // head-1 partial att_out [64][32]
#define SMEM_FLT  (OFF_ATT1 + 2048)

__global__ __launch_bounds__(NTHR)
void gga_fused_kernel(const float* __restrict__ x,
                      const float* __restrict__ nmean, const float* __restrict__ nstd,
                      const float* __restrict__ w1, const float* __restrict__ b1,
                      const float* __restrict__ w2, const float* __restrict__ b2,
                      const float* __restrict__ encw, const float* __restrict__ encb,
                      const float* __restrict__ srcw, const float* __restrict__ srcb,
                      const float* __restrict__ dstw, const float* __restrict__ dstb,
                      const float* __restrict__ decw, const float* __restrict__ decb,
                      float* __restrict__ out)
{
  extern __shared__ float sm[];
  const int tid = threadIdx.x;
  const int wv  = tid >> 5;          // wave id 0..3
  const int ln  = tid & 31;          // lane
  const int lr  = ln & 15;           // fragment row-lane
  const int lh  = ln >> 4;           // lane half (selects K=2,3 / M=8..15)

  // ================= stage weights into LDS (fragment-major for B operands) =================
  // W1 -> B-fragments: frag[(k*8+n)*64 + l*2 + j] = W1[4k + 2*(l>>4) + j][n*16 + (l&15)]
  for (int i = tid; i < 8192; i += NTHR) {
    int kn = i >> 6, k = kn >> 3, n = kn & 7;
    int l = (i & 63) >> 1, j = i & 1;
    sm[OFF_W1 + i] = w1[(4 * k + 2 * (l >> 4) + j) * H1D + n * 16 + (l & 15)];
  }
  // W2 -> B-fragments
  for (int i = tid; i < 4096; i += NTHR) {
    int kn = i >> 6, k = kn >> 1, n = kn & 1;
    int l = (i & 63) >> 1, j = i & 1;
    sm[OFF_W2 + i] = w2[(4 * k + 2 * (l >> 4) + j) * H2D + n * 16 + (l & 15)];
  }
  // Encoder: fold (x-mean)/(std+1e-8) into a block-diagonal [64,32] weight, frag-major.
  for (int i = tid; i < 2048; i += NTHR) {
    int kn = i >> 6, k = kn >> 1, n = kn & 1;
    int l = (i & 63) >> 1, j = i & 1;
    int row = 4 * k + 2 * (l >> 4) + j;      // 0..63 (input feature)
    int col = n * 16 + (l & 15);             // 0..31 (g*4+e)
    int grow = row >> 3, f = row & 7, gcol = col >> 2, e = col & 3;
    float is = 1.0f / (nstd[row] + 1e-8f);
    sm[OFF_ENCW + i] = (grow == gcol) ? encw[grow * 32 + f * 4 + e] * is : 0.0f;
  }
  if (tid < H1D) sm[OFF_B1 + tid] = b1[tid];
  if (tid < H2D) sm[OFF_B2 + tid] = b2[tid];
  if (tid < 32) {                            // folded encoder bias
    int g = tid >> 2, e = tid & 3;
    float acc = encb[tid];
    #pragma unroll
    for (int f = 0; f < 8; ++f) {
      int r = g * 8 + f;
      acc -= nmean[r] * (1.0f / (nstd[r] + 1e-8f)) * encw[g * 32 + f * 4 + e];
    }
    sm[OFF_ENCB + tid] = acc;
  }
  const float iva = 0.35355339059327373f;    // 1/sqrt(A): folded into src projection
  if (tid < 64) {                            // src/dst weight fragments [4,16]
    int l = tid >> 1, j = tid & 1;
    int row = 2 * (l >> 4) + j, col = l & 15;
    sm[OFF_SRCW + tid] = srcw[row * 16 + col] * iva;
    sm[OFF_DSTW + tid] = dstw[row * 16 + col];
  }
  if (tid < 16) { sm[OFF_SRCB + tid] = srcb[tid] * iva; sm[OFF_DSTB + tid] = dstb[tid]; }
  for (int i = tid; i < 192; i += NTHR) sm[OFF_DECW + i] = decw[i];
  if (tid < 2) sm[OFF_DECB + tid] = decb[tid];
  __syncthreads();

  // ================= per-wave pointers =================
  float* slab  = sm + OFF_SLAB + wv * SLAB_W;
  float* hid1s = slab;               // [16][132]  (dead after GEMM2)
  float* srcs  = slab;               // [128][17]  (reuses hid1 space)
  float* dsts  = slab + 2176;        // [128][17]
  float* xhats = sm + OFF_XHAT + wv * 544;   // [16][34]
  float* hs    = sm + OFF_H    + wv * 544;   // [16][34]

  const long long base = (long long)blockIdx.x * SPB + wv * 16;

  // A-fragments of x (16 K-steps of 4), straight from global: lane lr holds row, lh selects K pair.
  v2f ax[16];
  {
    const float* xrow = x + (size_t)(base + lr) * DIN + 2 * lh;
    #pragma unroll
    for (int k = 0; k < 16; ++k) ax[k] = *(const v2f*)(xrow + 4 * k);
  }

  // ---------- GEMM1: hid1 = relu(x @ W1 + b1)  [16,64]x[64,128] ----------
  #pragma unroll
  for (int n = 0; n < 8; ++n) {
    v8f acc = {};
    #pragma unroll
    for (int k = 0; k < 16; ++k) {
      v2f bf = *(const v2f*)(sm + OFF_W1 + (k * 8 + n) * 64 + ln * 2);
      acc = WMMA4(ax[k], bf, acc);
    }
    int col = n * 16 + lr;
    float bias = sm[OFF_B1 + col];
    #pragma unroll
    for (int v = 0; v < 8; ++v) {
      float val = acc[v] + bias;
      hid1s[(v + 8 * lh) * 132 + col] = val > 0.0f ? val : 0.0f;
    }
  }

  // ---------- GEMM2: xhat = relu(hid1 @ W2 + b2)  [16,128]x[128,32] ----------
  {
    v8f a0 = {}, a1 = {};
    #pragma unroll
    for (int k = 0; k < 32; ++k) {
      v2f af = *(const v2f*)(hid1s + lr * 132 + 4 * k + 2 * lh);
      v2f b0 = *(const v2f*)(sm + OFF_W2 + (k * 2 + 0) * 64 + ln * 2);
      v2f b1f = *(const v2f*)(sm + OFF_W2 + (k * 2 + 1) * 64 + ln * 2);
      a0 = WMMA4(af, b0, a0);
      a1 = WMMA4(af, b1f, a1);
    }
    float c0 = sm[OFF_B2 + lr], c1 = sm[OFF_B2 + 16 + lr];
    #pragma unroll
    for (int v = 0; v < 8; ++v) {
      int row = v + 8 * lh;
      float v0 = a0[v] + c0, v1 = a1[v] + c1;
      xhats[row * 34 + lr]      = v0 > 0.0f ? v0 : 0.0f;
      xhats[row * 34 + 16 + lr] = v1 > 0.0f ? v1 : 0.0f;
    }
  }

  // ---------- GEMM3: h = relu(x @ ENCfold + bfold)  [16,64]x[64,32] ----------
  {
    v8f a0 = {}, a1 = {};
    #pragma unroll
    for (int k = 0; k < 16; ++k) {
      v2f b0 = *(const v2f*)(sm + OFF_ENCW + (k * 2 + 0) * 64 + ln * 2);
      v2f b1f = *(const v2f*)(sm + OFF_ENCW + (k * 2 + 1) * 64 + ln * 2);
      a0 = WMMA4(ax[k], b0, a0);
      a1 = WMMA4(ax[k], b1f, a1);
    }
    float c0 = sm[OFF_ENCB + lr], c1 = sm[OFF_ENCB + 16 + lr];
    #pragma unroll
    for (int v = 0; v < 8; ++v) {
      int row = v + 8 * lh;
      float v0 = a0[v] + c0, v1 = a1[v] + c1;
      hs[row * 34 + lr]      = v0 > 0.0f ? v0 : 0.0f;
      hs[row * 34 + 16 + lr] = v1 > 0.0f ? v1 : 0.0f;
    }
  }

  // ---------- GEMM4/5: src,dst = h2d @ {src_w,dst_w} + b  where h2d = [16*8, 4], N=16, K=4 ----------
  {
    v2f bsrc = *(const v2f*)(sm + OFF_SRCW + ln * 2);
    v2f bdst = *(const v2f*)(sm + OFF_DSTW + ln * 2);
    float sb = sm[OFF_SRCB + lr], db = sm[OFF_DSTB + lr];
    #pragma unroll
    for (int m = 0; m < 8; ++m) {
      int fr = m * 16 + lr;                  // flat (sample,group) row
      v2f af = *(const v2f*)(hs + (fr >> 3) * 34 + (fr & 7) * 4 + 2 * lh);
      v8f as = {}, ad = {};
      as = WMMA4(af, bsrc, as);
      ad = WMMA4(af, bdst, ad);
      #pragma unroll
      for (int v = 0; v < 8; ++v) {
        int row = m * 16 + v + 8 * lh;
        srcs[row * 17 + lr] = as[v] + sb;
        dsts[row * 17 + lr] = ad[v] + db;
      }
    }
  }
  __syncthreads();

  // ================= per-(sample, head) attention in VALU =================
  const int s  = tid & 63;                   // local sample
  const int hh = tid >> 6;                   // head 0/1
  const int sw = s >> 4, srow = s & 15;
  const float* hrow = sm + OFF_H + sw * 544 + srow * 34;           // h[s]  (32 floats, g*4+e)
  const float* sp   = sm + OFF_SLAB + sw * SLAB_W + (srow * 8) * 17;          // src rows
  const float* dp   = sm + OFF_SLAB + sw * SLAB_W + 2176 + (srow * 8) * 17;   // dst rows

  float hreg[32];
  #pragma unroll
  for (int i = 0; i < 32; ++i) hreg[i] = hrow[i];
  float dr[64];
  #pragma unroll
  for (int k = 0; k < 8; ++k)
    #pragma unroll
    for (int a = 0; a < 8; ++a) dr[k * 8 + a] = dp[k * 17 + hh * 8 + a];

  float atth[32];
  #pragma unroll
  for (int i = 0; i < 32; ++i) atth[i] = 0.0f;

  #pragma unroll
  for (int g = 0; g < 8; ++g) {
    float sg[8];
    #pragma unroll
    for (int a = 0; a < 8; ++a) sg[a] = sp[g * 17 + hh * 8 + a];   // already /sqrt(A)
    float sc[8], mx = -3.4e38f;
    #pragma unroll
    for (int k = 0; k < 8; ++k) {
      float d = 0.0f;
      #pragma unroll
      for (int a = 0; a < 8; ++a) d += sg[a] * dr[k * 8 + a];
      sc[k] = d;
      mx = fmaxf(mx, d);
    }
    float sum = 0.0f;
    #pragma unroll
    for (int k = 0; k < 8; ++k) { sc[k] = __expf(sc[k] - mx); sum += sc[k]; }
    float inv = 1.0f / sum;
    #pragma unroll
    for (int k = 0; k < 8; ++k) {
      float w = sc[k] * inv;
      #pragma unroll
      for (int e = 0; e < 4; ++e) atth[g * 4 + e] += w * hreg[k * 4 + e];
    }
  }
  if (hh == 1) {
    float* ab = sm + OFF_ATT1 + s * 32;
    #pragma unroll
    for (int i = 0; i < 32; ++i) ab[i] = atth[i];
  }
  __syncthreads();

  // ================= decoder (head-0 thread per sample) =================
  if (hh == 0) {
    const float* xh = sm + OFF_XHAT + sw * 544 + srow * 34;
    const float* a1 = sm + OFF_ATT1 + s * 32;
    float o0 = sm[OFF_DECB + 0], o1 = sm[OFF_DECB + 1];
    #pragma unroll
    for (int i = 0; i < 32; ++i) {
      float f = xh[i];
      o0 += f * sm[OFF_DECW + i * 2 + 0];
      o1 += f * sm[OFF_DECW + i * 2 + 1];
    }
    #pragma unroll
    for (int i = 0; i < 32; ++i) {
      float f = hreg[i];
      o0 += f * sm[OFF_DECW + (32 + i) * 2 + 0];
      o1 += f * sm[OFF_DECW + (32 + i) * 2 + 1];
    }
    #pragma unroll
    for (int i = 0; i < 32; ++i) {
      float f = atth[i] + a1[i];
      o0 += f * sm[OFF_DECW + (64 + i) * 2 + 0];
      o1 += f * sm[OFF_DECW + (64 + i) * 2 + 1];
    }
    size_t gs = (size_t)blockIdx.x * SPB + s;
    out[gs * 2 + 0] = o0;
    out[gs * 2 + 1] = o1;
  }
}

extern "C" void kernel_launch(void* const* d_in, const int* in_sizes, int n_in,
                              void* d_out, int out_size, void* d_ws, size_t ws_size,
                              hipStream_t stream) {
  const float* x     = (const float*)d_in[0];
  const float* nmean = (const float*)d_in[1];
  const float* nstd  = (const float*)d_in[2];
  const float* w1    = (const float*)d_in[3];
  const float* b1    = (const float*)d_in[4];
  const float* w2    = (const float*)d_in[5];
  const float* b2    = (const float*)d_in[6];
  const float* encw  = (const float*)d_in[7];
  const float* encb  = (const float*)d_in[8];
  const float* srcw  = (const float*)d_in[9];
  const float* srcb  = (const float*)d_in[10];
  const float* dstw  = (const float*)d_in[11];
  const float* dstb  = (const float*)d_in[12];
  const float* decw  = (const float*)d_in[13];
  const float* decb  = (const float*)d_in[14];
  float* out = (float*)d_out;

  const size_t smem = (size_t)SMEM_FLT * sizeof(float);  // ~151 KB; 320 KB/WGP on CDNA5
  hipFuncSetAttribute((const void*)gga_fused_kernel,
                      hipFuncAttributeMaxDynamicSharedMemorySize, (int)smem);
  gga_fused_kernel<<<NBLK, NTHR, smem, stream>>>(
      x, nmean, nstd, w1, b1, w2, b2, encw, encb,
      srcw, srcb, dstw, dstb, decw, decb, out);
}